// Model_72490458021992
// MI455X (gfx1250) — compile-verified
//
#include <hip/hip_runtime.h>
#include <hip/hip_bf16.h>

// ---------------------------------------------------------------------------
// Soft-silhouette rasterization loss for MI455X (gfx1250, wave32, WMMA).
// Edge-function evaluation runs on the matrix pipe via V_WMMA_F32_16X16X4_F32:
//   D(16 faces x 16 pixels) = A(16x4 edge coeffs) x B(4x16 pixel coords)
// Transcendentals: log1p(-sigmoid(z)) == -softplus(z), evaluated with raw
// v_exp_f32 / v_log_f32 (base-2 builtins, constants folded) to avoid the
// libm denorm fix-up sequences.
// ---------------------------------------------------------------------------

typedef __attribute__((ext_vector_type(2))) float v2f;
typedef __attribute__((ext_vector_type(8))) float v8f;

#define IMGW 256
#define NFACE 1024
#define NVERT 1026
#define INV_SIGMA 100.0f          /* 1/0.01 */
#define SP_MAX 13.815510558f      /* -log(1e-6), from cov clip at 1-1e-6 */
#define EPSF 1e-8f
#define LOG2E 1.4426950408889634f
#define LN2 0.6931471805599453f

// ------------------------- kernel 0: zero the scalar output ----------------
__global__ void zero_out_kernel(float* out) { out[0] = 0.0f; }

// ------------------------- kernel 1: vertex transform ----------------------
__global__ void vert_kernel(const float* __restrict__ verts,
                            float* __restrict__ xy, int V) {
  int i = blockIdx.x * blockDim.x + threadIdx.x;
  if (i >= V) return;
  // eye at dist=2.732, elev=0, azim=90deg  (numpy double trig, then f32)
  const float ex = 2.732f, ey = 0.0f, ez = -2.732f * 6.123234e-17f;
  // z = normalize(at - eye)
  float zx = -ex, zy = -ey, zz = -ez;
  float zn = sqrtf(zx * zx + zy * zy + zz * zz) + EPSF;
  zx /= zn; zy /= zn; zz /= zn;
  // x = normalize(cross(up=(0,1,0), z)) = (zz, 0, -zx)/|.|
  float xx = zz, xy_ = 0.0f, xz = -zx;
  float xn = sqrtf(xx * xx + xy_ * xy_ + xz * xz) + EPSF;
  xx /= xn; xy_ /= xn; xz /= xn;
  // y = normalize(cross(z, x))
  float yx = zy * xz - zz * xy_;
  float yy = zz * xx - zx * xz;
  float yz = zx * xy_ - zy * xx;
  float yn = sqrtf(yx * yx + yy * yy + yz * yz) + EPSF;
  yx /= yn; yy /= yn; yz /= yn;

  float vx = verts[i * 3 + 0] - ex;
  float vy = verts[i * 3 + 1] - ey;
  float vz = verts[i * 3 + 2] - ez;
  float cx = xx * vx + xy_ * vy + xz * vz;
  float cy = yx * vx + yy * vy + yz * vz;
  float cz = zx * vx + zy * vy + zz * vz;
  const float w = 0.5773502691896258f;  // tan(30 deg)
  float denom = cz * w + EPSF;
  xy[i * 2 + 0] = cx / denom;
  xy[i * 2 + 1] = cy / denom;
}

// ------------------------- kernel 2: per-face edge coefficients ------------
// coef[f][e][0..2] = (-ey, ex, ey*ax - ex*ay) / area   so w = A*X + B*Y + C
__global__ void face_kernel(const int* __restrict__ faces,
                            const float* __restrict__ xy,
                            float* __restrict__ coef, int F) {
  int f = blockIdx.x * blockDim.x + threadIdx.x;
  if (f >= F) return;
  int i0 = faces[f * 3 + 0];
  int i1 = faces[f * 3 + 1];
  int i2 = faces[f * 3 + 2];
  float x0 = xy[i0 * 2], y0 = xy[i0 * 2 + 1];
  float x1 = xy[i1 * 2], y1 = xy[i1 * 2 + 1];
  float x2 = xy[i2 * 2], y2 = xy[i2 * 2 + 1];
  float area = (x1 - x0) * (y2 - y0) - (y1 - y0) * (x2 - x0);
  if (fabsf(area) < EPSF) area = EPSF;
  float inv = 1.0f / area;
  float axv[3] = {x0, x1, x2}, ayv[3] = {y0, y1, y2};
  float bxv[3] = {x1, x2, x0}, byv[3] = {y1, y2, y0};
#pragma unroll
  for (int e = 0; e < 3; ++e) {
    float eyv = byv[e] - ayv[e];
    float exv = bxv[e] - axv[e];
    coef[f * 9 + e * 3 + 0] = -eyv * inv;
    coef[f * 9 + e * 3 + 1] =  exv * inv;
    coef[f * 9 + e * 3 + 2] = (eyv * axv[e] - exv * ayv[e]) * inv;
  }
}

// ------------------------- kernel 3: main loss (WMMA) ----------------------
// 8 waves/block; each wave owns 16 pixels of one row; loops 1024 faces in
// chunks of 16 using three V_WMMA_F32_16X16X4_F32 (one per triangle edge).
__global__ __launch_bounds__(256) void sil_loss_kernel(
    const float* __restrict__ coef, const float* __restrict__ image_ref,
    float* __restrict__ out) {
  __shared__ float sC[NFACE * 9];  // 36 KB of the 320 KB WGP LDS
  for (int i = threadIdx.x; i < NFACE * 9; i += 256) sC[i] = coef[i];
  __syncthreads();

  const int lane   = threadIdx.x & 31;
  const int waveId = threadIdx.x >> 5;
  const int gw     = blockIdx.x * 8 + waveId;   // 4096 waves total
  const int row    = gw >> 4;                   // image row
  const int col    = ((gw & 15) << 4) + (lane & 15);
  const bool hi    = lane >= 16;

  const float step = 2.0f / (IMGW - 1);
  const float Xp = -1.0f + step * (float)col;
  const float Yp = -1.0f + step * (float)row;

  // B(4x16): lanes 0-15 carry rows K=0,1 = (X, Y); lanes 16-31 rows K=2,3 = (1, 0)
  v2f b;
  b.x = hi ? 1.0f : Xp;
  b.y = hi ? 0.0f : Yp;
  const v8f cz = {};  // zero accumulator

  float S = 0.0f;  // sum over faces of min(softplus(wmin/sigma), SP_MAX)
  for (int f0 = 0; f0 < NFACE; f0 += 16) {
    // Uniform (non-divergent) LDS loads: every lane reads all 9 coefficients
    // of its face, then role-selects in registers. Keeps EXEC all-ones for
    // the WMMAs and lets the compiler issue wide ds_load_b96s.
    const float* fc = &sC[(f0 + (lane & 15)) * 9];
    float c0 = fc[0], c1 = fc[1], c2 = fc[2];
    float c3 = fc[3], c4 = fc[4], c5 = fc[5];
    float c6 = fc[6], c7 = fc[7], c8 = fc[8];

    // A(16x4): lanes 0-15 carry K=0,1 = (Acoef, Bcoef); lanes 16-31 K=2,3 = (Ccoef, 0)
    v2f a0, a1, a2;
    a0.x = hi ? c2 : c0;  a0.y = hi ? 0.0f : c1;
    a1.x = hi ? c5 : c3;  a1.y = hi ? 0.0f : c4;
    a2.x = hi ? c8 : c6;  a2.y = hi ? 0.0f : c7;

    v8f w0 = __builtin_amdgcn_wmma_f32_16x16x4_f32(false, a0, false, b,
                                                   (short)0, cz, false, false);
    v8f w1 = __builtin_amdgcn_wmma_f32_16x16x4_f32(false, a1, false, b,
                                                   (short)0, cz, false, false);
    v8f w2 = __builtin_amdgcn_wmma_f32_16x16x4_f32(false, a2, false, b,
                                                   (short)0, cz, false, false);
#pragma unroll
    for (int k = 0; k < 8; ++k) {
      float wmin = fminf(w0[k], fminf(w1[k], w2[k]));   // -> v_min3_num_f32
      float z = wmin * INV_SIGMA;
      // log(1 - clip(sigmoid(z),0,1-1e-6)) = -min(softplus(z), SP_MAX)
      // softplus via raw hw exp2/log2: no libm denorm fix-up sequences.
      float t  = __builtin_amdgcn_exp2f(-fabsf(z) * LOG2E);   // v_exp_f32
      float sp = fmaxf(z, 0.0f) + LN2 * __builtin_amdgcn_logf(1.0f + t);  // v_log_f32
      S += fminf(sp, SP_MAX);
    }
  }

  // Rows 0-7 live in lanes 0-15, rows 8-15 in lanes 16-31: merge halves.
  S += __shfl_xor(S, 16, 32);

  float val = 0.0f;
  if (!hi) {
    float sil = 1.0f - __builtin_amdgcn_exp2f(-S * LOG2E);
    float d = sil - image_ref[row * IMGW + col];
    val = d * d;
  }
  // wave32 butterfly reduction, then one float atomic per wave
  val += __shfl_xor(val, 1, 32);
  val += __shfl_xor(val, 2, 32);
  val += __shfl_xor(val, 4, 32);
  val += __shfl_xor(val, 8, 32);
  val += __shfl_xor(val, 16, 32);
  if (lane == 0) atomicAdd(out, val);
}

// ---------------------------------------------------------------------------
extern "C" void kernel_launch(void* const* d_in, const int* in_sizes, int n_in,
                              void* d_out, int out_size, void* d_ws,
                              size_t ws_size, hipStream_t stream) {
  (void)in_sizes; (void)n_in; (void)out_size; (void)ws_size;
  const float* verts     = (const float*)d_in[0];  // (1,1026,3) f32
  const float* image_ref = (const float*)d_in[1];  // (1,256,256) f32
  const int*   faces     = (const int*)d_in[2];    // (1,1024,3) int

  float* out  = (float*)d_out;
  float* xy   = (float*)d_ws;                       // 1026*2 f32 = 8.2 KB
  float* coef = (float*)((char*)d_ws + 16384);      // 1024*9 f32 = 36 KB

  zero_out_kernel<<<1, 1, 0, stream>>>(out);
  vert_kernel<<<(NVERT + 127) / 128, 128, 0, stream>>>(verts, xy, NVERT);
  face_kernel<<<(NFACE + 127) / 128, 128, 0, stream>>>(faces, xy, coef, NFACE);
  sil_loss_kernel<<<512, 256, 0, stream>>>(coef, image_ref, out);
}